// Interaction_24756191494484
// MI455X (gfx1250) — compile-verified
//
#include <hip/hip_runtime.h>
#include <cstdint>

// ================= problem constants =================
#define BB       8
#define PP       96
#define HH       96
#define DD       128
#define H_CHUNK  6                 // h-tiles per workgroup; grid = (96, 96/6 = 16)

// LDS tile: 128 rows x 128 f32; TDM pads +4 dwords per 128-dword row
// -> row stride 132 dwords: A-fragment ds reads are 64-bank conflict-free.
#define ROW_DW    132
#define BUF_DW    (ROW_DW * 128)        // 16896 dwords
#define BUF_BYTES (BUF_DW * 4)          // 67584 B
#define SH_BYTES  (2 * BUF_BYTES + 256) // 2 tile buffers + partials scratch

typedef float    v2f __attribute__((ext_vector_type(2)));
typedef float    v4f __attribute__((ext_vector_type(4)));
typedef float    v8f __attribute__((ext_vector_type(8)));
typedef uint32_t v4u __attribute__((ext_vector_type(4)));
typedef int      v4i __attribute__((ext_vector_type(4)));
typedef int      v8i __attribute__((ext_vector_type(8)));

// ---- Tensor Data Mover: DMA one contiguous 64KB (p,h) tile -> LDS, with
// 4-dword padding after every 128 dwords (pad_interval=6 -> 512B, pad_amount=3 -> 4 dwords).
__device__ __forceinline__ void tdm_load_tile(uint64_t gaddr, uint32_t lds_off) {
  v4u g0;
  g0[0] = 1u;                                                    // count=1 (user descriptor)
  g0[1] = lds_off;                                               // lds_addr [63:32]
  g0[2] = (uint32_t)(gaddr & 0xffffffffu);                       // global_addr lo
  g0[3] = (uint32_t)((gaddr >> 32) & 0x01ffffffu) | (2u << 30);  // global_addr hi | type=2
  v8i g1;
  g1[0] = (int)((2u << 16) | (1u << 20) | (6u << 22) | (3u << 25)); // data_size=4B, pad_en, 128dw interval, +4dw
  g1[1] = (int)(16384u << 16);   // tensor_dim0 = 16384 (low 16)
  g1[2] = (int)(1u << 16);       // tensor_dim1 = 1
  g1[3] = (int)(16384u << 16);   // tile_dim0   = 16384
  g1[4] = (int)(1u | (1u << 16));// tile_dim1 = 1, tile_dim2 = 1
  g1[5] = 16384;                 // tensor_dim0_stride (lo 32)
  g1[6] = (int)(16384u << 16);   // tensor_dim1_stride (lo 16)
  g1[7] = 0;
  v4i g2 = {1, 0, 0, 0};         // tensor_dim2 = 1
  v4i g3 = {0, 0, 0, 0};
  v8i g4 = {0, 0, 0, 0, 0, 0, 0, 0};  // extra group required by 6-arg builtin (unused/zero)
  __builtin_amdgcn_tensor_load_to_lds(g0, g1, g2, g3, g4, 0);
}

// out[b,p,h] = sum_{i,j} premise[b,p,i] * kern[p,h,i,j] * hyp[b,h,j]
// Per (p,h) tile: T[i,b] = sum_j K[i,j]*hyp[b,j] via V_WMMA_F32_16X16X4_F32
// (A = K rows from LDS, B = hyp cols b padded 8->16), then dot rows of T with premise.
__global__ __launch_bounds__(128, 1) void interaction_kernel(
    const float* __restrict__ premise,
    const float* __restrict__ hyp,
    const float* __restrict__ kern,
    float* __restrict__ out)
{
  extern __shared__ char smem_raw[];
  float* lds0     = (float*)smem_raw;
  float* partials = (float*)(smem_raw + 2 * BUF_BYTES);  // [b(8)][wave*2+half(8)]

  const int p    = blockIdx.x;                 // 0..95
  const int h0   = blockIdx.y * H_CHUNK;       // chunk of 6 h values
  const int tid  = threadIdx.x;
  const int wave = tid >> 5;                   // 4 waves
  const int lane = tid & 31;
  const int m    = lane & 15;                  // A-row (i) / B-col (b) within 16-tile
  const int half = lane >> 4;                  // K-offset half per f32 WMMA layout
  const int bb   = m & 7;                      // clamped batch (lanes b>=8 are padding)

  // premise fragments: fixed for whole block (same p), reused for all 6 h tiles.
  // lane holds premise[b][p][i0..i0+7] and [i0+16..i0+23] matching C-fragment rows.
  const float* prow = premise + ((size_t)bb * PP + p) * DD;
  const int i0 = wave * 32 + half * 8;
  v4f pr0a = *(const v4f*)(prow + i0);
  v4f pr0b = *(const v4f*)(prow + i0 + 4);
  v4f pr1a = *(const v4f*)(prow + i0 + 16);
  v4f pr1b = *(const v4f*)(prow + i0 + 20);

  const uint32_t lds_base = (uint32_t)(uintptr_t)(void*)smem_raw;  // low 32 bits = LDS offset
  const uint64_t kbase = (uint64_t)(uintptr_t)kern +
                         ((uint64_t)p * HH + (uint64_t)h0) * (uint64_t)(DD * DD * 4);

  if (wave == 0) tdm_load_tile(kbase, lds_base);   // prologue: tile 0 -> buf 0

  for (int t = 0; t < H_CHUNK; ++t) {
    const int h = h0 + t;
    if (wave == 0) {
      if (t + 1 < H_CHUNK)                          // prefetch next tile into other buffer
        tdm_load_tile(kbase + (uint64_t)(t + 1) * (uint64_t)(DD * DD * 4),
                      lds_base + (uint32_t)((t + 1) & 1) * BUF_BYTES);
      if (t > 0 && lane < 8) {                      // write previous tile's result
        float s = 0.f;
        #pragma unroll
        for (int q = 0; q < 8; ++q) s += partials[lane * 8 + q];
        out[((size_t)lane * PP + p) * HH + (h - 1)] = s;
      }
      if (t + 1 < H_CHUNK) __builtin_amdgcn_s_wait_tensorcnt(1);  // tile t done (in-order)
      else                 __builtin_amdgcn_s_wait_tensorcnt(0);
    }
    __syncthreads();

    const float* buf   = lds0 + (size_t)(t & 1) * BUF_DW;
    const float* hrow  = hyp + ((size_t)bb * HH + h) * DD;
    const float* arow0 = buf + (size_t)(wave * 32 + m) * ROW_DW;   // wave owns i-tiles {2w, 2w+1}
    const float* arow1 = arow0 + 16 * ROW_DW;

    v8f c0 = {0,0,0,0,0,0,0,0};
    v8f c1 = {0,0,0,0,0,0,0,0};
    int jo = half * 2;                       // f32 WMMA A/B layout: lanes 16-31 hold K+2,K+3
    #pragma unroll
    for (int kj = 0; kj < 32; ++kj, jo += 4) {
      v2f bf = *(const v2f*)(hrow + jo);     // B-frag: hyp[b][j], hyp[b][j+1]
      v2f a0, a1;                            // A-frags from LDS (conflict-free banks)
      a0.x = arow0[jo]; a0.y = arow0[jo + 1];
      a1.x = arow1[jo]; a1.y = arow1[jo + 1];
      c0 = __builtin_amdgcn_wmma_f32_16x16x4_f32(false, a0, false, bf, (short)0, c0, false, false);
      c1 = __builtin_amdgcn_wmma_f32_16x16x4_f32(false, a1, false, bf, (short)0, c1, false, false);
    }

    // epilogue: part = sum_i premise[b,p,i] * T[i,b] for this wave's 32 i rows
    float part = 0.f;
    #pragma unroll
    for (int r = 0; r < 4; ++r) {
      part += pr0a[r] * c0[r] + pr0b[r] * c0[r + 4];
      part += pr1a[r] * c1[r] + pr1b[r] * c1[r + 4];
    }
    if (m < 8) partials[m * 8 + wave * 2 + half] = part;
    __syncthreads();
  }

  if (wave == 0 && lane < 8) {               // flush last tile
    float s = 0.f;
    #pragma unroll
    for (int q = 0; q < 8; ++q) s += partials[lane * 8 + q];
    out[((size_t)lane * PP + p) * HH + (h0 + H_CHUNK - 1)] = s;
  }
}

extern "C" void kernel_launch(void* const* d_in, const int* in_sizes, int n_in,
                              void* d_out, int out_size, void* d_ws, size_t ws_size,
                              hipStream_t stream) {
  (void)in_sizes; (void)n_in; (void)d_ws; (void)ws_size; (void)out_size;
  const float* premise = (const float*)d_in[0];
  const float* hyp     = (const float*)d_in[1];
  const float* kern    = (const float*)d_in[2];
  float* out           = (float*)d_out;

  // 135 KB dynamic LDS per workgroup (2 WGs / 320KB WGP) — opt in every call (deterministic).
  (void)hipFuncSetAttribute((const void*)interaction_kernel,
                            hipFuncAttributeMaxDynamicSharedMemorySize, SH_BYTES);

  dim3 grid(PP, HH / H_CHUNK);   // (96, 16)
  dim3 block(128);               // 4 waves
  interaction_kernel<<<grid, block, SH_BYTES, stream>>>(premise, hyp, kern, out);
}